// KGATModel_36155034698013
// MI455X (gfx1250) — compile-verified
//
#include <hip/hip_runtime.h>

#define NUM_USERS_C 50000
#define NNODES     200000
#define NSTRIPS    (NNODES / 16)   // 12500 row-strips of 16
#define OUT_STRIDE 176
#define SLOPE      0.01f

typedef float v2f __attribute__((ext_vector_type(2)));
typedef float v8f __attribute__((ext_vector_type(8)));

// ---------------------------------------------------------------------------
// ego = concat(user_emb, entity_emb);  out[:, 0:64] = ego (unnormalized)
// ---------------------------------------------------------------------------
__global__ void kgat_init_kernel(const float* __restrict__ user_emb,
                                 const float* __restrict__ ent_emb,
                                 float* __restrict__ ego,
                                 float* __restrict__ out) {
    int tid = blockIdx.x * blockDim.x + threadIdx.x;
    if (tid >= NNODES * 64) return;
    int row = tid >> 6;
    int c   = tid & 63;
    float v = (row < NUM_USERS_C) ? user_emb[tid]
                                  : ent_emb[tid - NUM_USERS_C * 64];
    ego[tid] = v;
    out[(size_t)row * OUT_STRIDE + c] = v;
}

__global__ void kgat_zero_kernel(float* __restrict__ p, int n) {
    int tid = blockIdx.x * blockDim.x + threadIdx.x;
    if (tid < n) p[tid] = 0.0f;
}

// ---------------------------------------------------------------------------
// side[r, :] += vals[e] * ego[c, :]   (atomic scatter-add; 16 threads/edge
// at din=64, float4 per thread -> coalesced gather; both the gathered matrix
// (51 MB) and the scatter target fit in the 192 MB L2, so atomics stay on-die)
// ---------------------------------------------------------------------------
__global__ void kgat_spmm_kernel(const int* __restrict__ rows,
                                 const int* __restrict__ cols,
                                 const float* __restrict__ vals,
                                 const float* __restrict__ x,
                                 float* __restrict__ y,
                                 int din, int nedges) {
    int chunks = din >> 2;
    int tid = blockIdx.x * blockDim.x + threadIdx.x;
    if (tid >= nedges * chunks) return;
    int e  = tid / chunks;
    int ch = tid - e * chunks;
    int r = rows[e];
    int c = cols[e];
    float v = vals[e];
    const float4 xv = *(const float4*)(x + (size_t)c * din + ch * 4);
    float* yp = y + (size_t)r * din + ch * 4;
    atomicAdd(yp + 0, v * xv.x);
    atomicAdd(yp + 1, v * xv.y);
    atomicAdd(yp + 2, v * xv.z);
    atomicAdd(yp + 3, v * xv.w);
}

// ---------------------------------------------------------------------------
// Bi-interaction aggregator:
//   ego_new = leaky((ego+side)@W1 + b1) + leaky((ego*side)@W2 + b2)
//
// One wave32 per 16-row strip; the wave owns the full 16 x DOUT output with
// NT = DOUT/16 accumulator pairs, so the A strip (ego/side rows) is read from
// global exactly once. W1/W2 are staged TRANSPOSED into LDS once per block
// (row stride DIN+4 -> the 32-lane ds_load_b64 covers all 64 banks exactly
// once, conflict-free), then every B fragment is an aligned LDS b64 read.
// V_WMMA_F32_16X16X4_F32 keeps exact f32 accumulation (reference parity);
// per k-step the wave issues 2*NT independent WMMAs back-to-back.
// ---------------------------------------------------------------------------
template <int DIN, int DOUT>
__global__ __launch_bounds__(256) void
kgat_gemm_bi_kernel(const float* __restrict__ ego,
                    const float* __restrict__ side,
                    const float* __restrict__ W1,
                    const float* __restrict__ B1,
                    const float* __restrict__ W2,
                    const float* __restrict__ B2,
                    float* __restrict__ ego_new) {
    constexpr int NT  = DOUT / 16;   // N-tiles per strip
    constexpr int WTS = DIN + 4;     // padded LDS row stride (bank-conflict-free)

    __shared__ float w1s[DOUT * WTS];
    __shared__ float w2s[DOUT * WTS];

    // cooperative transposed staging: w*s[n*WTS + k] = W[k*DOUT + n]
    for (int idx = threadIdx.x; idx < DIN * DOUT; idx += 256) {
        int k = idx / DOUT;
        int n = idx - k * DOUT;        // reads coalesced in n
        w1s[n * WTS + k] = W1[idx];
        w2s[n * WTS + k] = W2[idx];
    }
    __syncthreads();

    int lane  = threadIdx.x & 31;
    int wave  = threadIdx.x >> 5;
    int strip = blockIdx.x * 8 + wave;
    bool active = (strip < NSTRIPS);
    int s = active ? strip : 0;        // clamp: keep EXEC all-1s for WMMA

    int row0 = s << 4;
    int hl   = lane >> 4;              // lane half selects K[1] (ISA 7.12.2)
    int l16  = lane & 15;

    const float* pe = ego  + (size_t)(row0 + l16) * DIN;
    const float* ps = side + (size_t)(row0 + l16) * DIN;

    const v8f vzero = {0.f, 0.f, 0.f, 0.f, 0.f, 0.f, 0.f, 0.f};
    v8f c1[NT], c2[NT];
#pragma unroll
    for (int j = 0; j < NT; ++j) { c1[j] = vzero; c2[j] = vzero; }

    for (int k = 0; k < DIN; k += 4) {
        int kk = k + 2 * hl;
        v2f ev = *(const v2f*)(pe + kk);       // 8B-aligned global b64
        v2f sv = *(const v2f*)(ps + kk);
        v2f a1 = ev + sv;                      // (ego + side) fragment
        v2f a2 = ev * sv;                      // (ego * side) fragment
#pragma unroll
        for (int j = 0; j < NT; ++j) {
            v2f b1 = *(const v2f*)(&w1s[(j * 16 + l16) * WTS + kk]);
            v2f b2 = *(const v2f*)(&w2s[(j * 16 + l16) * WTS + kk]);
            c1[j] = __builtin_amdgcn_wmma_f32_16x16x4_f32(
                        false, a1, false, b1, (short)0, c1[j], false, false);
            c2[j] = __builtin_amdgcn_wmma_f32_16x16x4_f32(
                        false, a2, false, b2, (short)0, c2[j], false, false);
        }
    }

    if (active) {
#pragma unroll
        for (int j = 0; j < NT; ++j) {
            float bi1 = B1[j * 16 + l16];
            float bi2 = B2[j * 16 + l16];
#pragma unroll
            for (int v = 0; v < 8; ++v) {
                int m = v + 8 * hl;            // C/D: M = v + 8*half, N = lane&15
                float h1 = c1[j][v] + bi1;
                float h2 = c2[j][v] + bi2;
                float o = (h1 > 0.0f ? h1 : h1 * SLOPE) +
                          (h2 > 0.0f ? h2 : h2 * SLOPE);
                ego_new[(size_t)(row0 + m) * DOUT + j * 16 + l16] = o;
            }
        }
    }
}

// ---------------------------------------------------------------------------
// Per-row L2 normalize into output column slice; one wave32 per row,
// butterfly reduction over 32 lanes (wave32-safe).
// ---------------------------------------------------------------------------
__global__ void kgat_l2norm_kernel(const float* __restrict__ x,
                                   float* __restrict__ out,
                                   int dout, int col_off) {
    int lane = threadIdx.x & 31;
    int wave = threadIdx.x >> 5;
    int row = blockIdx.x * 8 + wave;
    if (row >= NNODES) return;
    const float* xr = x + (size_t)row * dout;
    float v0 = (lane < dout) ? xr[lane] : 0.0f;
    float v1 = (lane + 32 < dout) ? xr[lane + 32] : 0.0f;
    float s = v0 * v0 + v1 * v1;
#pragma unroll
    for (int off = 16; off > 0; off >>= 1)
        s += __shfl_xor(s, off, 32);
    float inv = 1.0f / fmaxf(sqrtf(s), 1e-12f);
    float* orow = out + (size_t)row * OUT_STRIDE + col_off;
    if (lane < dout)      orow[lane]      = v0 * inv;
    if (lane + 32 < dout) orow[lane + 32] = v1 * inv;
}

// ---------------------------------------------------------------------------
extern "C" void kernel_launch(void* const* d_in, const int* in_sizes, int n_in,
                              void* d_out, int out_size, void* d_ws, size_t ws_size,
                              hipStream_t stream) {
    const int*   rows     = (const int*)d_in[0];
    const int*   cols     = (const int*)d_in[1];
    const float* vals     = (const float*)d_in[2];
    const float* user_emb = (const float*)d_in[3];
    const float* ent_emb  = (const float*)d_in[4];
    const float* W1[3] = {(const float*)d_in[5],  (const float*)d_in[9],  (const float*)d_in[13]};
    const float* B1[3] = {(const float*)d_in[6],  (const float*)d_in[10], (const float*)d_in[14]};
    const float* W2[3] = {(const float*)d_in[7],  (const float*)d_in[11], (const float*)d_in[15]};
    const float* B2[3] = {(const float*)d_in[8],  (const float*)d_in[12], (const float*)d_in[16]};
    float* out = (float*)d_out;
    const int nedges = in_sizes[0];

    // workspace: ego / side / ego_new (N*64 f32 each; ping-pong ego<->ego_new)
    float* bufA = (float*)d_ws;
    float* bufB = bufA + (size_t)NNODES * 64;
    float* bufC = bufB + (size_t)NNODES * 64;

    {
        int n = NNODES * 64;
        kgat_init_kernel<<<(n + 255) / 256, 256, 0, stream>>>(user_emb, ent_emb, bufA, out);
    }

    const int dims[4] = {64, 64, 32, 16};
    float* ego     = bufA;
    float* side    = bufB;
    float* ego_new = bufC;
    int col_off = 64;
    const int gemm_blocks = (NSTRIPS + 7) / 8;

    for (int l = 0; l < 3; ++l) {
        int din  = dims[l];
        int dout = dims[l + 1];

        {
            int n = NNODES * din;
            kgat_zero_kernel<<<(n + 255) / 256, 256, 0, stream>>>(side, n);
        }
        {
            int nt = nedges * (din / 4);
            kgat_spmm_kernel<<<(nt + 255) / 256, 256, 0, stream>>>(
                rows, cols, vals, ego, side, din, nedges);
        }
        if (l == 0) {
            kgat_gemm_bi_kernel<64, 64><<<gemm_blocks, 256, 0, stream>>>(
                ego, side, W1[l], B1[l], W2[l], B2[l], ego_new);
        } else if (l == 1) {
            kgat_gemm_bi_kernel<64, 32><<<gemm_blocks, 256, 0, stream>>>(
                ego, side, W1[l], B1[l], W2[l], B2[l], ego_new);
        } else {
            kgat_gemm_bi_kernel<32, 16><<<gemm_blocks, 256, 0, stream>>>(
                ego, side, W1[l], B1[l], W2[l], B2[l], ego_new);
        }
        {
            kgat_l2norm_kernel<<<(NNODES + 7) / 8, 256, 0, stream>>>(
                ego_new, out, dout, col_off);
        }

        float* t = ego; ego = ego_new; ego_new = t;
        col_off += dout;
    }
}